// PoseGAT_13013750907607
// MI455X (gfx1250) — compile-verified
//
#include <hip/hip_runtime.h>
#include <hip/hip_bf16.h>

typedef __attribute__((ext_vector_type(2))) float v2f;
typedef __attribute__((ext_vector_type(8))) float v8f;

#define N_NODES 50000
#define N_GRAPH 2000
#define PER_G   25
#define N_EDGE  400000
#define HEADS   2
#define SLOPE   0.2f

// ---------------------------------------------------------------------------
// Utility: fill a float buffer with a constant
// ---------------------------------------------------------------------------
__global__ void fill_f32(float* __restrict__ p, float v, int n) {
    int i = blockIdx.x * blockDim.x + threadIdx.x;
    if (i < n) p[i] = v;
}

// ---------------------------------------------------------------------------
// fp32 GEMM via V_WMMA_F32_16X16X4_F32.
// C[Nrows x M] = A[Nrows x K] * W[K x M], all row-major.
// One 16x16 output tile per wave32. Requires Nrows%16==0, K%4==0, M%16==0.
// Fragment layouts per CDNA5 ISA 7.12.2:
//   A (16x4):  lane l -> row m = l&15;  a[v] = A[m][k + v + 2*(l>>4)]
//   B (4x16):  lane l -> col n = l&15;  b[v] = W[k + v + 2*(l>>4)][n]
//   C (16x16): lane l -> col n = l&15;  c[v] = C[v + 8*(l>>4)][n]
// ---------------------------------------------------------------------------
__global__ void gemm_f32_wmma(const float* __restrict__ A,
                              const float* __restrict__ W,
                              float* __restrict__ Cout,
                              int K, int M, int totalTiles) {
    const int wavesPerBlock = blockDim.x >> 5;
    int wave = blockIdx.x * wavesPerBlock + (threadIdx.x >> 5);
    if (wave >= totalTiles) return;          // wave-uniform guard (EXEC stays all-1)
    const int tilesN = M >> 4;
    const int tm = wave / tilesN;
    const int tn = wave % tilesN;
    const int lane = threadIdx.x & 31;
    const int hi = lane >> 4;                // 0 for lanes 0-15, 1 for lanes 16-31
    const int lo = lane & 15;

    const float* arow = A + (size_t)(tm * 16 + lo) * K + 2 * hi;
    const float* bcol = W + (size_t)(2 * hi) * M + tn * 16 + lo;

    v8f acc = {};
    for (int k = 0; k < K; k += 4) {
        v2f a, b;
        a.x = arow[k];
        a.y = arow[k + 1];
        b.x = bcol[(size_t)k * M];
        b.y = bcol[(size_t)(k + 1) * M];
        acc = __builtin_amdgcn_wmma_f32_16x16x4_f32(
            /*neg_a=*/false, a, /*neg_b=*/false, b,
            /*c_mod=*/(short)0, acc, /*reuse_a=*/false, /*reuse_b=*/false);
    }

    float* crow = Cout + (size_t)(tm * 16 + 8 * hi) * M + tn * 16 + lo;
#pragma unroll
    for (int v = 0; v < 8; ++v) crow[(size_t)v * M] = acc[v];
}

// ---------------------------------------------------------------------------
// Per (node, head): attention coefficients a_src, a_dst; init segment max/sum.
// hfeat layout: [N, HEADS*C] row-major, so (n,h) block starts at (n*HEADS+h)*C.
// ---------------------------------------------------------------------------
__global__ void attn_prep(const float* __restrict__ hfeat,
                          const float* __restrict__ att_src,
                          const float* __restrict__ att_dst,
                          float* __restrict__ asrc, float* __restrict__ adst,
                          float* __restrict__ mbuf, float* __restrict__ dbuf,
                          int C) {
    int i = blockIdx.x * blockDim.x + threadIdx.x;   // i = n*HEADS + h
    if (i >= N_NODES * HEADS) return;
    int h = i % HEADS;
    const float* hp = hfeat + (size_t)i * C;
    const float* as = att_src + h * C;
    const float* ad = att_dst + h * C;
    float s = 0.f, d = 0.f;
    for (int c = 0; c < C; ++c) {
        float v = hp[c];
        s += v * as[c];
        d += v * ad[c];
    }
    asrc[i] = s;
    adst[i] = d;
    mbuf[i] = -INFINITY;
    dbuf[i] = 0.f;
}

// Signed/unsigned bit-pattern trick for IEEE-754 atomic max (no NaNs present).
__device__ __forceinline__ void atomicMaxF(float* addr, float val) {
    if (val >= 0.f) atomicMax((int*)addr, __float_as_int(val));
    else            atomicMin((unsigned int*)addr, __float_as_uint(val));
}

// ---------------------------------------------------------------------------
// Edge pass 1: segment max of leaky_relu(a_src[s] + a_dst[d]) over destinations.
// Edges t in [0, E+N): t >= E are the implicit self-loops.
// ---------------------------------------------------------------------------
__global__ void edge_max(const int* __restrict__ src, const int* __restrict__ dst,
                         const float* __restrict__ asrc, const float* __restrict__ adst,
                         float* __restrict__ mbuf) {
    int t = blockIdx.x * blockDim.x + threadIdx.x;
    if (t >= N_EDGE + N_NODES) return;
    int s, d;
    if (t < N_EDGE) { s = src[t]; d = dst[t]; } else { s = d = t - N_EDGE; }
#pragma unroll
    for (int h = 0; h < HEADS; ++h) {
        float v = asrc[s * HEADS + h] + adst[d * HEADS + h];
        v = v > 0.f ? v : SLOPE * v;
        atomicMaxF(&mbuf[d * HEADS + h], v);
    }
}

// ---------------------------------------------------------------------------
// Edge pass 2: e = exp(val - max); store per-edge e; segment sum into denom.
// ---------------------------------------------------------------------------
__global__ void edge_exp(const int* __restrict__ src, const int* __restrict__ dst,
                         const float* __restrict__ asrc, const float* __restrict__ adst,
                         const float* __restrict__ mbuf, float* __restrict__ dbuf,
                         float* __restrict__ eexp) {
    int t = blockIdx.x * blockDim.x + threadIdx.x;
    if (t >= N_EDGE + N_NODES) return;
    int s, d;
    if (t < N_EDGE) { s = src[t]; d = dst[t]; } else { s = d = t - N_EDGE; }
#pragma unroll
    for (int h = 0; h < HEADS; ++h) {
        float v = asrc[s * HEADS + h] + adst[d * HEADS + h];
        v = v > 0.f ? v : SLOPE * v;
        float e = __expf(v - mbuf[d * HEADS + h]);
        eexp[t * HEADS + h] = e;
        atomicAdd(&dbuf[d * HEADS + h], e);
    }
}

// ---------------------------------------------------------------------------
// Edge pass 3: out[d,h,:] += alpha * h[s,h,:].  One thread per (edge,h,c).
// ---------------------------------------------------------------------------
__global__ void edge_scatter(const int* __restrict__ src, const int* __restrict__ dst,
                             const float* __restrict__ eexp, const float* __restrict__ dbuf,
                             const float* __restrict__ hfeat, float* __restrict__ accum,
                             int C, int total) {
    int gid = blockIdx.x * blockDim.x + threadIdx.x;
    if (gid >= total) return;
    int hc = HEADS * C;
    int t = gid / hc;
    int r = gid - t * hc;
    int h = r / C;
    int c = r - h * C;
    int s, d;
    if (t < N_EDGE) { s = src[t]; d = dst[t]; } else { s = d = t - N_EDGE; }
    float alpha = eexp[t * HEADS + h] / (dbuf[d * HEADS + h] + 1e-16f);
    atomicAdd(&accum[((size_t)d * HEADS + h) * C + c],
              alpha * hfeat[((size_t)s * HEADS + h) * C + c]);
}

// ---------------------------------------------------------------------------
// out = relu(accum + bias), per element over [N, HC].
// ---------------------------------------------------------------------------
__global__ void bias_relu(const float* __restrict__ accum, const float* __restrict__ bias,
                          float* __restrict__ out, int HC) {
    int i = blockIdx.x * blockDim.x + threadIdx.x;
    if (i >= N_NODES * HC) return;
    int j = i % HC;
    float v = accum[i] + bias[j];
    out[i] = v > 0.f ? v : 0.f;
}

// ---------------------------------------------------------------------------
// Layer 3 finalize: mean over heads (H=2, C3=32) + bias.
// ---------------------------------------------------------------------------
__global__ void head_mean_bias(const float* __restrict__ accum, const float* __restrict__ bias,
                               float* __restrict__ out) {
    int i = blockIdx.x * blockDim.x + threadIdx.x;   // over N * 32
    if (i >= N_NODES * 32) return;
    int n = i >> 5;
    int c = i & 31;
    out[i] = 0.5f * (accum[(size_t)n * 64 + c] + accum[(size_t)n * 64 + 32 + c]) + bias[c];
}

// ---------------------------------------------------------------------------
// global_mean_pool: 25 consecutive nodes per graph.
// ---------------------------------------------------------------------------
__global__ void mean_pool(const float* __restrict__ nf, float* __restrict__ out) {
    int i = blockIdx.x * blockDim.x + threadIdx.x;   // over G * 32
    if (i >= N_GRAPH * 32) return;
    int g = i >> 5;
    int c = i & 31;
    float s = 0.f;
    for (int k = 0; k < PER_G; ++k) s += nf[(size_t)(g * PER_G + k) * 32 + c];
    out[i] = s * (1.f / PER_G);
}

// ---------------------------------------------------------------------------
extern "C" void kernel_launch(void* const* d_in, const int* in_sizes, int n_in,
                              void* d_out, int out_size, void* d_ws, size_t ws_size,
                              hipStream_t stream) {
    const float* x   = (const float*)d_in[0];
    const int*   ei  = (const int*)d_in[1];
    // d_in[2] = batch (unused: node->graph mapping is n/25 by construction)
    const float* W1  = (const float*)d_in[3];
    const float* as1 = (const float*)d_in[4];
    const float* ad1 = (const float*)d_in[5];
    const float* b1  = (const float*)d_in[6];
    const float* W2  = (const float*)d_in[7];
    const float* as2 = (const float*)d_in[8];
    const float* ad2 = (const float*)d_in[9];
    const float* b2  = (const float*)d_in[10];
    const float* W3  = (const float*)d_in[11];
    const float* as3 = (const float*)d_in[12];
    const float* ad3 = (const float*)d_in[13];
    const float* b3  = (const float*)d_in[14];

    const int* src = ei;
    const int* dst = ei + N_EDGE;

    // workspace layout (floats)
    float* ws    = (float*)d_ws;
    float* buf1  = ws;                                  // N x 128  (GEMM out h)
    float* buf2  = buf1 + (size_t)N_NODES * 128;        // N x 128  (layer features)
    float* accum = buf2 + (size_t)N_NODES * 128;        // N x 128  (message accum)
    float* asrc  = accum + (size_t)N_NODES * 128;       // N x H
    float* adst  = asrc + (size_t)N_NODES * HEADS;      // N x H
    float* mbuf  = adst + (size_t)N_NODES * HEADS;      // N x H
    float* dbuf  = mbuf + (size_t)N_NODES * HEADS;      // N x H
    float* eexp  = dbuf + (size_t)N_NODES * HEADS;      // (E+N) x H
    float* buf3  = eexp + (size_t)(N_EDGE + N_NODES) * HEADS;  // N x 32

    const int EE = N_EDGE + N_NODES;
    const int B  = 256;

    // ---- helper lambda-free per-layer sequences ----
    // ================= Layer 1: in=x[.,64] -> feat buf2[.,128] =================
    {
        const int K = 64, M = 128, C = 64;
        int tiles = (N_NODES / 16) * (M / 16);
        gemm_f32_wmma<<<(tiles + 3) / 4, 128, 0, stream>>>(x, W1, buf1, K, M, tiles);
        attn_prep<<<(N_NODES * HEADS + B - 1) / B, B, 0, stream>>>(buf1, as1, ad1, asrc, adst, mbuf, dbuf, C);
        fill_f32<<<(N_NODES * M + B - 1) / B, B, 0, stream>>>(accum, 0.f, N_NODES * M);
        edge_max<<<(EE + B - 1) / B, B, 0, stream>>>(src, dst, asrc, adst, mbuf);
        edge_exp<<<(EE + B - 1) / B, B, 0, stream>>>(src, dst, asrc, adst, mbuf, dbuf, eexp);
        int total = EE * HEADS * C;
        edge_scatter<<<(total + B - 1) / B, B, 0, stream>>>(src, dst, eexp, dbuf, buf1, accum, C, total);
        bias_relu<<<(N_NODES * M + B - 1) / B, B, 0, stream>>>(accum, b1, buf2, M);
    }
    // ================= Layer 2: in=buf2[.,128] -> feat buf2[.,128] =================
    {
        const int K = 128, M = 128, C = 64;
        int tiles = (N_NODES / 16) * (M / 16);
        gemm_f32_wmma<<<(tiles + 3) / 4, 128, 0, stream>>>(buf2, W2, buf1, K, M, tiles);
        attn_prep<<<(N_NODES * HEADS + B - 1) / B, B, 0, stream>>>(buf1, as2, ad2, asrc, adst, mbuf, dbuf, C);
        fill_f32<<<(N_NODES * M + B - 1) / B, B, 0, stream>>>(accum, 0.f, N_NODES * M);
        edge_max<<<(EE + B - 1) / B, B, 0, stream>>>(src, dst, asrc, adst, mbuf);
        edge_exp<<<(EE + B - 1) / B, B, 0, stream>>>(src, dst, asrc, adst, mbuf, dbuf, eexp);
        int total = EE * HEADS * C;
        edge_scatter<<<(total + B - 1) / B, B, 0, stream>>>(src, dst, eexp, dbuf, buf1, accum, C, total);
        bias_relu<<<(N_NODES * M + B - 1) / B, B, 0, stream>>>(accum, b2, buf2, M);
    }
    // ================= Layer 3: in=buf2[.,128] -> pooled d_out[G,32] =================
    {
        const int K = 128, M = 64, C = 32;
        int tiles = (N_NODES / 16) * (M / 16);
        gemm_f32_wmma<<<(tiles + 3) / 4, 128, 0, stream>>>(buf2, W3, buf1, K, M, tiles);
        attn_prep<<<(N_NODES * HEADS + B - 1) / B, B, 0, stream>>>(buf1, as3, ad3, asrc, adst, mbuf, dbuf, C);
        fill_f32<<<(N_NODES * M + B - 1) / B, B, 0, stream>>>(accum, 0.f, N_NODES * M);
        edge_max<<<(EE + B - 1) / B, B, 0, stream>>>(src, dst, asrc, adst, mbuf);
        edge_exp<<<(EE + B - 1) / B, B, 0, stream>>>(src, dst, asrc, adst, mbuf, dbuf, eexp);
        int total = EE * HEADS * C;
        edge_scatter<<<(total + B - 1) / B, B, 0, stream>>>(src, dst, eexp, dbuf, buf1, accum, C, total);
        head_mean_bias<<<(N_NODES * 32 + B - 1) / B, B, 0, stream>>>(accum, b3, buf3);
        mean_pool<<<(N_GRAPH * 32 + B - 1) / B, B, 0, stream>>>(buf3, (float*)d_out);
    }
}